// Pooler_43250320671414
// MI455X (gfx1250) — compile-verified
//
#include <hip/hip_runtime.h>
#include <hip/hip_bf16.h>

typedef __attribute__((ext_vector_type(2))) float v2f;
typedef __attribute__((ext_vector_type(8))) float v8f;

#define PP    7
#define NS    14      // sample points per spatial dim (7 bins * 2 subsamples)
#define NBINS 49
#define NCH   256
#define CH_PER_PASS 128

// Bilinear sample with validity masking. Indices are pre-clamped so all 4 tap
// loads are always in-bounds; invalid samples are zeroed arithmetically (no
// divergent branch -> EXEC stays all-ones for the WMMA that follows).
__device__ __forceinline__ float bilin(const float* __restrict__ fc, int W,
                                       int sy, int sx,
                                       const int* sY0, const float* sFy, const float* sVy,
                                       const int* sX0, const float* sFx, const float* sVx)
{
    int   y0 = sY0[sy];
    float fy = sFy[sy];
    float my = sVy[sy];
    int   x0 = sX0[sx];
    float fx = sFx[sx];
    float mx = sVx[sx];
    const float* p = fc + (size_t)y0 * W + x0;
    float v00 = p[0];
    float v01 = p[1];
    float v10 = p[W];
    float v11 = p[W + 1];
    float hy = 1.0f - fy, hx = 1.0f - fx;
    float v  = hy * (hx * v00 + fx * v01) + fy * (hx * v10 + fx * v11);
    return v * my * mx;
}

__global__ __launch_bounds__(256)
void roi_pool_wmma(const float* __restrict__ boxes,
                   const int*   __restrict__ batch_ids,
                   const float* __restrict__ f0,
                   const float* __restrict__ f1,
                   const float* __restrict__ f2,
                   const float* __restrict__ f3,
                   float* __restrict__ out)
{
    __shared__ int   sY0[NS];
    __shared__ int   sX0[NS];
    __shared__ float sFy[NS];
    __shared__ float sFx[NS];
    __shared__ float sVy[NS];
    __shared__ float sVx[NS];
    // output staging tile: 128 channels x 49 bins, written scattered (conflict
    // free: 49*ln mod 64 distinct), read back coalesced as float4.
    __shared__ __align__(16) float sOut[CH_PER_PASS * NBINS];

    const int roi = blockIdx.x;

    // ---- per-ROI scalars (computed redundantly per thread; cheap) ----
    const float x1 = boxes[roi * 4 + 0];
    const float y1 = boxes[roi * 4 + 1];
    const float x2 = boxes[roi * 4 + 2];
    const float y2 = boxes[roi * 4 + 3];
    const int   b  = batch_ids[roi];

    // level mapping: floor(4 + log2(sqrt(area)/224 + eps)), clip [2,5] -> idx 0..3
    const float area = (x2 - x1) * (y2 - y1);
    const float sdim = sqrtf(area);
    const float lvlf = floorf(4.0f + log2f(sdim * (1.0f / 224.0f) + 1e-6f));
    const float lvlc = fminf(fmaxf(lvlf, 2.0f), 5.0f);
    const int   lvl  = (int)lvlc - 2;               // 0..3

    const float* feat = (lvl == 0) ? f0 : (lvl == 1) ? f1 : (lvl == 2) ? f2 : f3;
    const int   HW  = 200 >> lvl;                   // 200,100,50,25 (square)
    const float scl = 0.25f / (float)(1 << lvl);    // 0.25,0.125,0.0625,0.03125

    const float rx1 = x1 * scl;
    const float ry1 = y1 * scl;
    const float rw  = fmaxf(x2 * scl - rx1, 1.0f);
    const float rh  = fmaxf(y2 * scl - ry1, 1.0f);
    const float bw  = rw * (1.0f / PP);
    const float bh  = rh * (1.0f / PP);

    // ---- precompute 14 y-samples and 14 x-samples into LDS ----
    {
        int t = threadIdx.x;
        if (t < 2 * NS) {
            int  i   = (t < NS) ? t : (t - NS);
            bool isY = (t < NS);
            // grid[i] = (i>>1) + ((i&1)+0.5)/2
            float g = (float)(i >> 1) + ((float)(i & 1) + 0.5f) * 0.5f;
            float c = isY ? (ry1 + g * bh) : (rx1 + g * bw);
            float L = (float)HW;
            bool  valid = (c > -1.0f) && (c < L);
            c = fminf(fmaxf(c, 0.0f), L - 1.0f);
            int lo = (int)floorf(c);
            if (lo > HW - 2) lo = HW - 2;
            float fr = c - (float)lo;
            if (isY) { sY0[i] = lo; sFy[i] = fr; sVy[i] = valid ? 1.0f : 0.0f; }
            else     { sX0[i] = lo; sFx[i] = fr; sVx[i] = valid ? 1.0f : 0.0f; }
        }
    }
    __syncthreads();

    const int lane = threadIdx.x & 31;
    const int wave = threadIdx.x >> 5;   // 0..7
    const int ln   = lane & 15;          // column / channel-within-tile
    const int hh   = lane >> 4;          // half-wave: which 2 of the 4 bin samples

    const float* fbatch = feat + (size_t)b * NCH * HW * HW;

    // 8 waves x 16 channels = 128 channels per pass; 2 passes cover 256.
    for (int pass = 0; pass < 2; ++pass) {
        const int chLocal = wave * 16 + ln;                  // 0..127
        const int ch      = pass * CH_PER_PASS + chLocal;    // 0..255
        const float* fc = fbatch + (size_t)ch * HW * HW;

        // 4 accumulator tiles cover bin rows 0..63 (49 valid)
        #pragma unroll
        for (int tt = 0; tt < 4; ++tt) {
            v8f acc = {};
            const int bStart = tt * 16;
            const int bEnd   = (tt == 3) ? NBINS : (bStart + 16);
            for (int bin = bStart; bin < bEnd; ++bin) {
                const int by = bin / PP;
                const int bx = bin - by * PP;
                // this half-wave computes sample row (by*2 + hh), both x subsamples
                const int sy = by * 2 + hh;
                const int sx = bx * 2;
                float vA = bilin(fc, HW, sy, sx,     sY0, sFy, sVy, sX0, sFx, sVx);
                float vB = bilin(fc, HW, sy, sx + 1, sY0, sFy, sVy, sX0, sFx, sVx);

                // A: 0.25 on row (bin%16) across all 4 K columns (K-permutation safe)
                const float aval = (ln == (bin & 15)) ? 0.25f : 0.0f;
                v2f a; a.x = aval; a.y = aval;
                v2f bb; bb.x = vA; bb.y = vB;

                acc = __builtin_amdgcn_wmma_f32_16x16x4_f32(
                          /*neg_a=*/false, a, /*neg_b=*/false, bb,
                          /*c_mod=*/(short)0, acc,
                          /*reuse_a=*/false, /*reuse_b=*/false);
            }
            // D layout: VGPR r -> M=r (lanes 0-15), M=r+8 (lanes 16-31); N = lane&15
            // Scatter into LDS staging tile (conflict-free stride).
            #pragma unroll
            for (int r = 0; r < 8; ++r) {
                const int bin = tt * 16 + r + 8 * hh;
                if (bin < NBINS) {
                    sOut[chLocal * NBINS + bin] = acc[r];
                }
            }
        }
        __syncthreads();

        // Cooperative fully-coalesced store of the contiguous 128*49-float
        // region: each b128 store instruction writes 512 contiguous bytes.
        {
            float4* dst = (float4*)(out + ((size_t)roi * NCH + pass * CH_PER_PASS) * NBINS);
            const float4* src = (const float4*)sOut;
            const int nvec = (CH_PER_PASS * NBINS) / 4;   // 1568
            for (int i = threadIdx.x; i < nvec; i += 256) {
                dst[i] = src[i];
            }
        }
        __syncthreads();   // protect sOut before pass 2 overwrites it
    }
}

extern "C" void kernel_launch(void* const* d_in, const int* in_sizes, int n_in,
                              void* d_out, int out_size, void* d_ws, size_t ws_size,
                              hipStream_t stream)
{
    const float* boxes = (const float*)d_in[0];
    const int*   bids  = (const int*)d_in[1];
    const float* f0    = (const float*)d_in[2];
    const float* f1    = (const float*)d_in[3];
    const float* f2    = (const float*)d_in[4];
    const float* f3    = (const float*)d_in[5];
    float*       outp  = (float*)d_out;

    const int nroi = in_sizes[0] / 4;   // boxes is [N,4]
    roi_pool_wmma<<<nroi, 256, 0, stream>>>(boxes, bids, f0, f1, f2, f3, outp);
}